// N2ReferenceEmbedding_15530601742779
// MI455X (gfx1250) — compile-verified
//
#include <hip/hip_runtime.h>
#include <hip/hip_bf16.h>

typedef float v2f __attribute__((ext_vector_type(2)));
typedef float v8f __attribute__((ext_vector_type(8)));

#define D_MODEL 1024
#define NTILES  (D_MODEL / 16)   // 64 channel tiles

// Build the striped WMMA B-operand table directly from W[1024,12] and bias.
// out[s,d] = c2[d]*t^2 + c1[d]*t + c0[d], t = float(N_tokens[s]), where
//   c2 = W0+W3+W6+W9
//   c1 = W1+W2+2W4+2W8+2W10
//   c0 = W2+2W4-W5+W8+W10+bias
// B (4x16 f32) VGPR striping for V_WMMA_F32_16X16X4_F32:
//   lanes 0-15 : {K=0 row, K=1 row} = {c2[n], c1[n]}
//   lanes 16-31: {K=2 row, K=3 row} = {c0[n], 0}
// Bops[tile*32 + lane] is exactly the v2f that lane loads for that tile.
__global__ __launch_bounds__(256) void n2_bpack_kernel(
    const float* __restrict__ W, const float* __restrict__ bias,
    v2f* __restrict__ Bops) {
  int slot = blockIdx.x * blockDim.x + threadIdx.x;   // [0, 64*32)
  if (slot < NTILES * 32) {
    int lane = slot & 31;
    int tile = slot >> 5;
    int n = (tile << 4) + (lane & 15);                // channel index
    const float* w = W + n * 12;
    v2f b;
    if (lane < 16) {
      b.x = w[0] + w[3] + w[6] + w[9];                                  // c2
      b.y = w[1] + w[2] + 2.0f * w[4] + 2.0f * w[8] + 2.0f * w[10];     // c1
    } else {
      b.x = w[2] + 2.0f * w[4] - w[5] + w[8] + w[10] + bias[n];         // c0
      b.y = 0.0f;                                                       // K=3 pad
    }
    Bops[slot] = b;
  }
}

// Each wave: 16 tokens, loop over 64 channel tiles of 16.
// One V_WMMA_F32_16X16X4_F32 per 16x16 output tile:
//   A (16x4): lanes 0-15 -> {t^2, t} (K=0,1), lanes 16-31 -> {1, 0} (K=2,3)
//   D (16x16 f32): VGPR j -> row j + 8*(lane>=16), col = lane&15
__global__ __launch_bounds__(256) void n2_embed_wmma_kernel(
    const int* __restrict__ toks, const v2f* __restrict__ Bops,
    float* __restrict__ out) {
  const int lane = threadIdx.x & 31;
  const int wave = threadIdx.x >> 5;
  const int tokBase = (blockIdx.x * 8 + wave) * 16;
  const int m = lane & 15;

  const float t = (float)toks[tokBase + m];
  const bool lo = (lane < 16);

  v2f a;
  a.x = lo ? (t * t) : 1.0f;  // K=0 (lanes 0-15) / K=2 (lanes 16-31)
  a.y = lo ? t : 0.0f;        // K=1              / K=3

  const v2f* __restrict__ bptr = Bops + lane;
  const int rowOff = (lane >> 4) << 3;  // 0 or 8
  float* __restrict__ outBase = out + (size_t)tokBase * D_MODEL + m;

  for (int c = 0; c < NTILES; ++c) {
    v2f b = bptr[c << 5];  // coalesced b64 load, no selects, EXEC untouched

    v8f acc = {};
    acc = __builtin_amdgcn_wmma_f32_16x16x4_f32(
        /*neg_a=*/false, a, /*neg_b=*/false, b,
        /*c_mod=*/(short)0, acc, /*reuse_a=*/false, /*reuse_b=*/false);

    float* __restrict__ p = outBase + (c << 4);
#pragma unroll
    for (int j = 0; j < 8; ++j) {
      p[(size_t)(j + rowOff) * D_MODEL] = acc[j];
    }
  }
}

extern "C" void kernel_launch(void* const* d_in, const int* in_sizes, int n_in,
                              void* d_out, int out_size, void* d_ws, size_t ws_size,
                              hipStream_t stream) {
  const int* toks   = (const int*)d_in[0];    // N_tokens, flat [4*8192]
  const float* W    = (const float*)d_in[1];  // [1024,12]
  const float* bias = (const float*)d_in[2];  // [1024]
  float* out        = (float*)d_out;          // [4*8192*1024] f32

  v2f* Bops = (v2f*)d_ws;                     // 64*32 v2f = 16 KB

  n2_bpack_kernel<<<(NTILES * 32 + 255) / 256, 256, 0, stream>>>(W, bias, Bops);

  const int numTok = in_sizes[0];             // 32768
  const int blocks = numTok / (16 * 8);       // 8 waves/block, 16 tokens/wave
  n2_embed_wmma_kernel<<<blocks, 256, 0, stream>>>(toks, Bops, out);
}